// Scorer_28999619182927
// MI455X (gfx1250) — compile-verified
//
#include <hip/hip_runtime.h>
#include <hip/hip_bf16.h>

typedef float v2f __attribute__((ext_vector_type(2)));
typedef float v4f __attribute__((ext_vector_type(4)));
typedef float v8f __attribute__((ext_vector_type(8)));

#define Bdim 2
#define Sdim 512
#define Cdim 25
#define Hdim 768
#define Fdim 3072
#define LDTP 1536

// Bs pair layout: Bs2[k/2][n*2 + (k&1)], row stride PB (even, pad +2).
// -> B fragment (B[k][n], B[k+1][n]) is one aligned ds_load_b64.
#define PB 130

// ---------------------------------------------------------------------------
// f32 WMMA GEMM: C = A(MxK) * B(KxN) (+ bias[N]), row-major.
// 256 threads = 8 waves, block tile 64x64, wave tile 16x32 (2 accumulators
// share one A fragment), K chunk 32. All fragments load as ds_load_b64.
// ---------------------------------------------------------------------------
template <bool GUARD_M>
__global__ __launch_bounds__(256) void gemm_wmma_f32(
    const float* __restrict__ A, int lda,
    const float* __restrict__ Bm, int ldb,
    const float* __restrict__ bias,   // may be nullptr
    float* __restrict__ Cm, int ldc,
    int M, int N, int K)
{
    __shared__ float As[64 * 36];    // 64 x 32, stride 36 (x4 -> b128 stores OK)
    __shared__ float Bs[16 * PB];    // 32 x 64 in k-paired layout

    const int t    = threadIdx.x;
    const int wave = t >> 5;
    const int lane = t & 31;
    const int hi   = lane >> 4;
    const int l15  = lane & 15;
    const int wm   = wave >> 1;      // 0..3
    const int wn   = wave & 1;       // 0..1

    const int m0 = blockIdx.y * 64;
    const int n0 = blockIdx.x * 64;

    v8f acc0 = {};
    v8f acc1 = {};

    const int arow  = wm * 16 + l15;
    const int bcol0 = (wn * 32 + l15) * 2;

    for (int k0 = 0; k0 < K; k0 += 32) {
        __syncthreads();
        // A tile: 64x32 = 512 float4, 2 per thread
        #pragma unroll
        for (int i = 0; i < 2; ++i) {
            int idx = t + i * 256;
            int r = idx >> 3, c4 = (idx & 7) * 4;
            v4f v;
            if (GUARD_M && (m0 + r) >= M) v = (v4f){0.f, 0.f, 0.f, 0.f};
            else v = *(const v4f*)&A[(size_t)(m0 + r) * lda + k0 + c4];
            *(v4f*)&As[r * 36 + c4] = v;
        }
        // B tile: 32x64 = 512 float4, 2 per thread, scattered into pair layout
        #pragma unroll
        for (int i = 0; i < 2; ++i) {
            int idx = t + i * 256;
            int r = idx >> 4, c4 = (idx & 15) * 4;
            v4f v = *(const v4f*)&Bm[(size_t)(k0 + r) * ldb + n0 + c4];
            #pragma unroll
            for (int q = 0; q < 4; ++q)
                Bs[(r >> 1) * PB + (c4 + q) * 2 + (r & 1)] = v[q];
        }
        if (k0 + 32 < K)
            __builtin_prefetch(&Bm[(size_t)(k0 + 32 + (t >> 3)) * ldb + n0 + (t & 7) * 8], 0, 1);
        __syncthreads();

        #pragma unroll
        for (int kk = 0; kk < 32; kk += 4) {
            const int prow = (kk >> 1) + hi;            // pair row for this half
            v2f a  = *(const v2f*)&As[arow * 36 + kk + 2 * hi];
            v2f b0 = *(const v2f*)&Bs[prow * PB + bcol0];
            v2f b1 = *(const v2f*)&Bs[prow * PB + bcol0 + 32];
            acc0 = __builtin_amdgcn_wmma_f32_16x16x4_f32(
                false, a, false, b0, (short)0, acc0, false, false);
            acc1 = __builtin_amdgcn_wmma_f32_16x16x4_f32(
                false, a, false, b1, (short)0, acc1, false, false);
        }
    }

    const int cn0 = n0 + wn * 32 + l15;
    const float bv0 = bias ? bias[cn0] : 0.0f;
    const float bv1 = bias ? bias[cn0 + 16] : 0.0f;
    #pragma unroll
    for (int i = 0; i < 8; ++i) {
        int r = m0 + wm * 16 + i + 8 * hi;   // C/D: VGPR i -> row i (lo) / i+8 (hi)
        if (!GUARD_M || r < M) {
            Cm[(size_t)r * ldc + cn0]      = acc0[i] + bv0;
            Cm[(size_t)r * ldc + cn0 + 16] = acc1[i] + bv1;
        }
    }
}

// ---------------------------------------------------------------------------
// Fused einsum + relu + W2 kernel.
// Block handles (b, c, 64-row s-tile). The whole scaled tile
// A' = tb[s0:s0+64,:] * lb[b,c,:]  (64x768 f32 ~= 197 KB) stays resident in
// LDS (CDNA5: 320 KB/WGP). F swept in 64-wide tiles; W1p chunks are
// double-buffered (k-paired layout) with register pipelining.
// ---------------------------------------------------------------------------
#define ALD  (Hdim + 4)           // 772: multiple of 4 -> b128 stores aligned
#define NCHUNK (Hdim / 32)        // 24
#define BSZ (16 * PB)             // one Bs buffer (pair rows)
#define SMEM_FUSED ((64 * ALD + 2 * BSZ + 192) * sizeof(float))

__global__ __launch_bounds__(256) void fused_bilinear_out(
    const float* __restrict__ tp,    // (B*S, 1536): [:,768:] = tb
    const float* __restrict__ lp,    // (B*C, 1536): [:,768:] = lb
    const float* __restrict__ W1p,   // (768, 3072) = W1[2H:3H]
    const float* __restrict__ tA,    // (B*S, 3072) = ta @ W1t
    const float* __restrict__ lA,    // (B*C, 3072) = la @ W1l + b1
    const float* __restrict__ W2,    // (3072, 3)
    const float* __restrict__ b2,    // (3)
    float* __restrict__ out)         // (B, S, C, 3)
{
    extern __shared__ float smem[];
    float* As     = smem;                 // 64 * ALD
    float* Bsbuf  = As + 64 * ALD;        // 2 * BSZ
    float* outAcc = Bsbuf + 2 * BSZ;      // 64 * 3

    const int t    = threadIdx.x;
    const int wave = t >> 5;
    const int lane = t & 31;
    const int hi   = lane >> 4;
    const int l15  = lane & 15;
    const int wm   = wave >> 1;           // 0..3  (16-row sub-tile)
    const int wn   = wave & 1;            // 0..1  (32-col F half)

    const int b  = blockIdx.z;
    const int c  = blockIdx.y;
    const int s0 = blockIdx.x * 64;

    const float* tbRow = tp + ((size_t)(b * Sdim + s0)) * LDTP + Hdim;
    const float* lbRow = lp + ((size_t)(b * Cdim + c)) * LDTP + Hdim;
    const float* lARow = lA + ((size_t)(b * Cdim + c)) * Fdim;

    // ---- stage A' = tb .* lb into LDS (column-scaled), b128 traffic ----
    {
        const int tx = t & 63;            // 16B column block
        const int ty = t >> 6;            // row phase 0..3
        v4f lbq[3];
        #pragma unroll
        for (int j = 0; j < 3; ++j)
            lbq[j] = *(const v4f*)&lbRow[(tx + 64 * j) * 4];
        for (int i = 0; i < 16; ++i) {
            int r = ty + 4 * i;
            #pragma unroll
            for (int j = 0; j < 3; ++j) {
                int cc = (tx + 64 * j) * 4;
                v4f v = *(const v4f*)&tbRow[(size_t)r * LDTP + cc];
                *(v4f*)&As[r * ALD + cc] = v * lbq[j];
            }
        }
    }
    if (t < 192) outAcc[t] = 0.0f;
    __syncthreads();

    float o[8][3];
    #pragma unroll
    for (int i = 0; i < 8; ++i)
        #pragma unroll
        for (int j = 0; j < 3; ++j) o[i][j] = 0.0f;

    const int arow  = wm * 16 + l15;
    const int bcol0 = (wn * 32 + l15) * 2;
    const int ldr   = (t + 0) >> 4;       // staging k-row 0..15 (first half)
    const int ldc4  = (t & 15) * 4;       // staging 16B col block

    for (int f0 = 0; f0 < Fdim; f0 += 64) {
        v8f acc0 = {};
        v8f acc1 = {};

        // prologue: chunk 0 -> regs -> Bs[0]
        v4f rg[2];
        #pragma unroll
        for (int i = 0; i < 2; ++i) {
            int r = (t + i * 256) >> 4;
            rg[i] = *(const v4f*)&W1p[(size_t)r * Fdim + f0 + ldc4];
        }
        #pragma unroll
        for (int i = 0; i < 2; ++i) {
            int r = (t + i * 256) >> 4;
            #pragma unroll
            for (int q = 0; q < 4; ++q)
                Bsbuf[(r >> 1) * PB + (ldc4 + q) * 2 + (r & 1)] = rg[i][q];
        }
        __syncthreads();

        for (int ch = 0; ch < NCHUNK; ++ch) {
            float* Bs = Bsbuf + (ch & 1) * BSZ;
            // issue next chunk's global loads before computing this one
            if (ch + 1 < NCHUNK) {
                int kn = (ch + 1) * 32;
                #pragma unroll
                for (int i = 0; i < 2; ++i) {
                    int r = (t + i * 256) >> 4;
                    rg[i] = *(const v4f*)&W1p[(size_t)(kn + r) * Fdim + f0 + ldc4];
                }
            }
            const int k0 = ch * 32;
            #pragma unroll
            for (int kk = 0; kk < 32; kk += 4) {
                const int prow = (kk >> 1) + hi;
                v2f a  = *(const v2f*)&As[arow * ALD + k0 + kk + 2 * hi];
                v2f b0 = *(const v2f*)&Bs[prow * PB + bcol0];
                v2f b1 = *(const v2f*)&Bs[prow * PB + bcol0 + 32];
                acc0 = __builtin_amdgcn_wmma_f32_16x16x4_f32(
                    false, a, false, b0, (short)0, acc0, false, false);
                acc1 = __builtin_amdgcn_wmma_f32_16x16x4_f32(
                    false, a, false, b1, (short)0, acc1, false, false);
            }
            if (ch + 1 < NCHUNK) {
                float* Bn = Bsbuf + ((ch + 1) & 1) * BSZ;
                #pragma unroll
                for (int i = 0; i < 2; ++i) {
                    int r = (t + i * 256) >> 4;
                    #pragma unroll
                    for (int q = 0; q < 4; ++q)
                        Bn[(r >> 1) * PB + (ldc4 + q) * 2 + (r & 1)] = rg[i][q];
                }
                __syncthreads();
            }
        }

        // ---- epilogue for this 64-wide F tile (two 16-col sub-tiles) ----
        #pragma unroll
        for (int sub = 0; sub < 2; ++sub) {
            const v8f& acc = sub ? acc1 : acc0;
            const int f = f0 + wn * 32 + sub * 16 + l15;
            const float lAv = lARow[f];
            const float w20 = W2[f * 3 + 0];
            const float w21 = W2[f * 3 + 1];
            const float w22 = W2[f * 3 + 2];
            #pragma unroll
            for (int i = 0; i < 8; ++i) {
                int srow = s0 + wm * 16 + i + 8 * hi;
                float v = acc[i] + tA[(size_t)(b * Sdim + srow) * Fdim + f] + lAv;
                v = v > 0.0f ? v : 0.0f;
                o[i][0] += v * w20;
                o[i][1] += v * w21;
                o[i][2] += v * w22;
            }
        }
    }

    // reduce the 16 N-columns held across the lanes of each 16-lane half
    #pragma unroll
    for (int i = 0; i < 8; ++i)
        #pragma unroll
        for (int j = 0; j < 3; ++j)
            #pragma unroll
            for (int m = 1; m < 16; m <<= 1)
                o[i][j] += __shfl_xor(o[i][j], m, 32);

    if (l15 == 0) {
        #pragma unroll
        for (int i = 0; i < 8; ++i) {
            int rloc = wm * 16 + i + 8 * hi;
            #pragma unroll
            for (int j = 0; j < 3; ++j)
                atomicAdd(&outAcc[rloc * 3 + j], o[i][j]);
        }
    }
    __syncthreads();
    if (t < 192) {
        int r = t / 3, j = t - r * 3;
        int s = s0 + r;
        out[(((size_t)b * Sdim + s) * Cdim + c) * 3 + j] = outAcc[t] + b2[j];
    }
}

// ---------------------------------------------------------------------------
extern "C" void kernel_launch(void* const* d_in, const int* in_sizes, int n_in,
                              void* d_out, int out_size, void* d_ws, size_t ws_size,
                              hipStream_t stream) {
    const float* token = (const float*)d_in[0];   // (B,S,H)
    const float* label = (const float*)d_in[1];   // (B,C,H)
    const float* Wt    = (const float*)d_in[2];   // (H,2H)
    const float* bt    = (const float*)d_in[3];   // (2H)
    const float* Wl    = (const float*)d_in[4];   // (H,2H)
    const float* bl    = (const float*)d_in[5];   // (2H)
    const float* W1    = (const float*)d_in[6];   // (3H,4H)
    const float* b1    = (const float*)d_in[7];   // (4H)
    const float* W2    = (const float*)d_in[8];   // (4H,3)
    const float* b2    = (const float*)d_in[9];   // (3)
    float* out = (float*)d_out;

    // workspace carve-up (floats): ~19.8 MB total
    float* tp  = (float*)d_ws;                       // 1024 x 1536
    float* lpd = tp  + (size_t)(Bdim * Sdim) * LDTP; //   50 x 1536
    float* tA  = lpd + (size_t)(Bdim * Cdim) * LDTP; // 1024 x 3072
    float* lA  = tA  + (size_t)(Bdim * Sdim) * Fdim; //   50 x 3072

    const int Mt = Bdim * Sdim;   // 1024 (multiple of 64 -> unguarded)
    const int Ml = Bdim * Cdim;   // 50   (guarded)

    // 1) tp = token @ Wt + bt
    gemm_wmma_f32<false><<<dim3(LDTP / 64, Mt / 64), 256, 0, stream>>>(
        token, Hdim, Wt, LDTP, bt, tp, LDTP, Mt, LDTP, Hdim);

    // 2) lp = label @ Wl + bl
    gemm_wmma_f32<true><<<dim3(LDTP / 64, 1), 256, 0, stream>>>(
        label, Hdim, Wl, LDTP, bl, lpd, LDTP, Ml, LDTP, Hdim);

    // 3) tA = ta @ W1t   (ta = tp[:, :768], W1t = W1[0:768])
    gemm_wmma_f32<false><<<dim3(Fdim / 64, Mt / 64), 256, 0, stream>>>(
        tp, LDTP, W1, Fdim, nullptr, tA, Fdim, Mt, Fdim, Hdim);

    // 4) lA = la @ W1l + b1   (la = lp[:, :768], W1l = W1[768:1536])
    gemm_wmma_f32<true><<<dim3(Fdim / 64, 1), 256, 0, stream>>>(
        lpd, LDTP, W1 + (size_t)Hdim * Fdim, Fdim, b1, lA, Fdim, Ml, Fdim, Hdim);

    // 5) fused einsum + relu + W2  (W1p = W1[1536:2304])
    (void)hipFuncSetAttribute((const void*)fused_bilinear_out,
                              hipFuncAttributeMaxDynamicSharedMemorySize,
                              (int)SMEM_FUSED);
    fused_bilinear_out<<<dim3(Sdim / 64, Cdim, Bdim), 256, SMEM_FUSED, stream>>>(
        tp, lpd, W1 + (size_t)(2 * Hdim) * Fdim, tA, lA, W2, b2, out);
}